// SqErr_cov_24945170055793
// MI455X (gfx1250) — compile-verified
//
#include <hip/hip_runtime.h>

typedef float v2f __attribute__((ext_vector_type(2)));
typedef float v8f __attribute__((ext_vector_type(8)));

#define NN   64
#define LDSS 65   // padded LDS row stride (floats): bank = (r + j) % 64, conflict-free columns

// One wave32 per batch: blocked Cholesky (16x16 blocks, WMMA f32 16x16x4 trailing
// updates), then forward substitution L y = x, partial = sum(log Lii) + 0.5*||y||^2.
__global__ __launch_bounds__(32) void chol_nll_kernel(const float* __restrict__ x,
                                                      const float* __restrict__ cov,
                                                      float* __restrict__ part)
{
    __shared__ float S[NN * LDSS];
    __shared__ float r[NN];

    const int lane = threadIdx.x;      // 0..31
    const int b    = blockIdx.x;
    const int hi16 = lane >> 4;        // 0 or 1
    const int l16  = lane & 15;

    // ---- stream cov[b] (64x64 fp32) into padded LDS, coalesced b128 loads ----
    const float4* src = (const float4*)(cov + (size_t)b * NN * NN);
    #pragma unroll
    for (int t = 0; t < (NN * NN / 4) / 32; ++t) {   // 32 iterations
        const int e4 = t * 32 + lane;                // float4 index 0..1023
        float4 v = src[e4];
        const int row = (e4 * 4) >> 6;
        const int col = (e4 * 4) & 63;
        float* dst = &S[row * LDSS + col];
        dst[0] = v.x; dst[1] = v.y; dst[2] = v.z; dst[3] = v.w;
    }
    r[lane]      = x[(size_t)b * NN + lane];
    r[lane + 32] = x[(size_t)b * NN + lane + 32];
    __syncthreads();

    float ldacc = 0.0f;   // accumulates sum(log Lii)  (== 0.5*logdet)

    #pragma unroll
    for (int k = 0; k < 4; ++k) {
        const int bk = 16 * k;

        // --- 1. factor diagonal block (right-looking; lanes 0..15 own rows) ---
        for (int j = 0; j < 16; ++j) {
            const float djj = S[(bk + j) * LDSS + (bk + j)];   // broadcast read
            const float ljj = __builtin_sqrtf(djj);
            ldacc += __builtin_logf(ljj);                       // same on all lanes
            const float inv = 1.0f / ljj;
            if (lane == j) S[(bk + j) * LDSS + (bk + j)] = ljj;
            if (lane < 16 && lane > j)
                S[(bk + lane) * LDSS + (bk + j)] *= inv;
            __syncthreads();
            if (lane < 16 && lane > j) {
                const float lij = S[(bk + lane) * LDSS + (bk + j)];
                for (int c = j + 1; c <= lane; ++c)
                    S[(bk + lane) * LDSS + (bk + c)] -= lij * S[(bk + c) * LDSS + (bk + j)];
            }
            __syncthreads();
        }

        if (k < 3) {
            // --- 2. panel solve: rows below diag block, X <- X * Lkk^{-T} ---
            const int nrows = (3 - k) * 16;
            for (int rr = lane; rr < nrows; rr += 32) {
                const int row = bk + 16 + rr;
                float v[16];
                #pragma unroll
                for (int j = 0; j < 16; ++j) v[j] = S[row * LDSS + bk + j];
                #pragma unroll
                for (int j = 0; j < 16; ++j) {
                    float t = v[j];
                    #pragma unroll
                    for (int u = 0; u < j; ++u)
                        t -= v[u] * S[(bk + j) * LDSS + bk + u];   // Lkk[j][u]
                    v[j] = t / S[(bk + j) * LDSS + bk + j];
                }
                #pragma unroll
                for (int j = 0; j < 16; ++j) S[row * LDSS + bk + j] = v[j];
            }
            __syncthreads();

            // --- 3. trailing update T(bi,bj) -= P_bi * P_bj^T via WMMA f32 16x16x4 ---
            // A operand (16x4): a[r]  = P_bi[l16][kk0 + r]
            // B operand (4x16): b[r]  = (P_bj^T)[kk0+r][l16] = P_bj[l16][kk0 + r]
            // C/D (16x16):      c[v]  = C[v + 8*hi16][l16]
            for (int bi = k + 1; bi < 4; ++bi) {
                for (int bj = k + 1; bj <= bi; ++bj) {
                    v8f acc = {0.f, 0.f, 0.f, 0.f, 0.f, 0.f, 0.f, 0.f};
                    #pragma unroll
                    for (int c = 0; c < 4; ++c) {
                        const int kk = bk + 4 * c + 2 * hi16;
                        v2f a, bb;
                        a.x  = S[(16 * bi + l16) * LDSS + kk];
                        a.y  = S[(16 * bi + l16) * LDSS + kk + 1];
                        bb.x = S[(16 * bj + l16) * LDSS + kk];
                        bb.y = S[(16 * bj + l16) * LDSS + kk + 1];
                        acc = __builtin_amdgcn_wmma_f32_16x16x4_f32(
                                false, a, false, bb, (short)0, acc, false, false);
                    }
                    #pragma unroll
                    for (int vv = 0; vv < 8; ++vv)
                        S[(16 * bi + vv + 8 * hi16) * LDSS + 16 * bj + l16] -= acc[vv];
                }
            }
            __syncthreads();
        }
    }

    // ---- forward substitution: y = L^{-1} x, quad = ||y||^2 ----
    float qacc = 0.0f;
    for (int i = 0; i < NN; ++i) {
        const float yi = r[i] / S[i * LDSS + i];   // broadcast; same on all lanes
        qacc += yi * yi;
        int row = i + 1 + lane;
        if (row < NN) r[row] -= S[row * LDSS + i] * yi;
        row += 32;
        if (row < NN) r[row] -= S[row * LDSS + i] * yi;
        __syncthreads();
    }

    if (lane == 0) part[b] = ldacc + 0.5f * qacc;
}

// Deterministic fixed-order reduction of per-batch partials.
__global__ __launch_bounds__(256) void reduce_kernel(const float* __restrict__ part,
                                                     float* __restrict__ out, int n)
{
    __shared__ float sm[256];
    const int tid = threadIdx.x;
    float s = 0.0f;
    for (int i = tid; i < n; i += 256) s += part[i];
    sm[tid] = s;
    __syncthreads();
    for (int off = 128; off > 0; off >>= 1) {
        if (tid < off) sm[tid] += sm[tid + off];
        __syncthreads();
    }
    if (tid == 0) out[0] = sm[0];
}

extern "C" void kernel_launch(void* const* d_in, const int* in_sizes, int n_in,
                              void* d_out, int out_size, void* d_ws, size_t ws_size,
                              hipStream_t stream)
{
    const float* x   = (const float*)d_in[0];   // [B, 64]
    const float* cov = (const float*)d_in[1];   // [B, 64, 64]
    const int B = in_sizes[0] / NN;             // 8192

    float* part = (float*)d_ws;                 // B floats of scratch

    chol_nll_kernel<<<B, 32, 0, stream>>>(x, cov, part);
    reduce_kernel<<<1, 256, 0, stream>>>(part, (float*)d_out, B);
}